// SD_layer_24567212933531
// MI455X (gfx1250) — compile-verified
//
#include <hip/hip_runtime.h>
#include <hip/hip_bf16.h>
#include <math.h>

#define N_NODES 20000
#define N_EDGES 320000
#define DIM     256
#define RDIM    200
#define NUM_ET  10
#define NB      6

#define ROWTILES  5                  // 16-row tiles per block
#define TILEROWS  (16 * ROWTILES)    // 80 rows/block; 20000 = 250 * 80
#define LDSSTRIDE 260                // padded row stride (bank-conflict free, 16B aligned)

typedef __attribute__((ext_vector_type(2))) float v2f;
typedef __attribute__((ext_vector_type(8))) float v8f;

// ---------------------------------------------------------------------------
// Kernel 1: fused GEMM  Wx = x @ W^T (which=0 -> ws)  and  res = x @ Wres^T
// (which=1 -> d_out).  v_wmma_f32_16x16x4_f32; 80-row x-tile in LDS so each
// B-fragment (W row pair) feeds 5 WMMAs.
// ---------------------------------------------------------------------------
struct GemmArgs {
    const float* x[NB];
    const float* W;      // [NB][256][256]
    const float* Wres;   // [NB][256][256]
    float*       Wx;     // ws, [NB][N][256]
    float*       out;    // d_out, [NB][N][256]
};

__global__ __launch_bounds__(256) void gemm_wmma_kernel(GemmArgs a) {
    extern __shared__ float xs[];              // TILEROWS x LDSSTRIDE
    const int rowTile = blockIdx.x;            // 0..249
    const int b       = blockIdx.y;            // 0..5
    const int which   = blockIdx.z;            // 0: W, 1: Wres
    const int tid     = threadIdx.x;
    const int lane    = tid & 31;
    const int wv      = tid >> 5;              // wave 0..7

    const float* x       = a.x[b];
    const int    rowBase = rowTile * TILEROWS;

    // stage x tile (coalesced float4 loads, 16B-aligned LDS stores)
    #pragma unroll
    for (int rep = 0; rep < TILEROWS / 4; ++rep) {
        int pos = rep * 1024 + tid * 4;        // float index in TILEROWSx256
        int r = pos >> 8;
        int c = pos & 255;
        float4 v = *(const float4*)(x + (size_t)(rowBase + r) * DIM + c);
        *(float4*)(&xs[r * LDSSTRIDE + c]) = v;
    }
    __syncthreads();

    const float* Wm  = (which == 0 ? a.W : a.Wres) + (size_t)b * DIM * DIM;
    float*       dst = (which == 0 ? a.Wx : a.out) + (size_t)b * N_NODES * DIM;

    const int m     = lane & 15;
    const int khalf = (lane & 16) ? 2 : 0;     // lanes 16-31 hold K+2,K+3
    const int col0  = (2 * wv) * 16 + m;
    const int col1  = (2 * wv + 1) * 16 + m;

    const float* wrow0 = Wm + (size_t)col0 * DIM + khalf;
    const float* wrow1 = Wm + (size_t)col1 * DIM + khalf;

    const float* aP[ROWTILES];
    #pragma unroll
    for (int t = 0; t < ROWTILES; ++t)
        aP[t] = &xs[(16 * t + m) * LDSSTRIDE + khalf];

    v8f acc[ROWTILES][2] = {};

    #pragma unroll 2
    for (int k0 = 0; k0 < DIM; k0 += 4) {
        v2f b0 = *(const v2f*)(wrow0 + k0);       // L2-resident W row frag
        v2f b1 = *(const v2f*)(wrow1 + k0);
        #pragma unroll
        for (int t = 0; t < ROWTILES; ++t) {
            v2f af = *(const v2f*)(aP[t] + k0);   // ds_load_b64
            acc[t][0] = __builtin_amdgcn_wmma_f32_16x16x4_f32(
                            false, af, false, b0, (short)0, acc[t][0], false, false);
            acc[t][1] = __builtin_amdgcn_wmma_f32_16x16x4_f32(
                            false, af, false, b1, (short)0, acc[t][1], false, false);
        }
    }

    const int rOff = (lane & 16) ? 8 : 0;      // C/D layout: vgpr v -> M=v / v+8
    #pragma unroll
    for (int t = 0; t < ROWTILES; ++t) {
        #pragma unroll
        for (int v = 0; v < 8; ++v) {
            size_t row = (size_t)(rowBase + 16 * t + v + rOff);
            dst[row * DIM + col0] = acc[t][0][v];
            dst[row * DIM + col1] = acc[t][1][v];
        }
    }
}

// ---------------------------------------------------------------------------
// Kernel 2: per-node scalars si = Wx . a_i, sj = Wx . a_j ; init m = -inf, s = 0
// ---------------------------------------------------------------------------
struct NodeArgs {
    const float* Wx;
    const float* A;      // [NB][768]
    float* si; float* sj; float* m; float* s;
};

__global__ __launch_bounds__(256) void node_scalar_kernel(NodeArgs a) {
    int g    = blockIdx.x * 8 + (threadIdx.x >> 5);   // global row over NB*N
    int lane = threadIdx.x & 31;
    if (g >= NB * N_NODES) return;
    int b = g / N_NODES;
    const float* row = a.Wx + (size_t)g * DIM;
    const float* ai  = a.A + (size_t)b * (3 * DIM);
    const float* aj  = ai + DIM;
    float di = 0.f, dj = 0.f;
    #pragma unroll
    for (int rep = 0; rep < 2; ++rep) {
        int c = rep * 128 + lane * 4;
        float4 v  = *(const float4*)(row + c);
        float4 wi = *(const float4*)(ai + c);
        float4 wj = *(const float4*)(aj + c);
        di += v.x * wi.x + v.y * wi.y + v.z * wi.z + v.w * wi.w;
        dj += v.x * wj.x + v.y * wj.y + v.z * wj.z + v.w * wj.w;
    }
    #pragma unroll
    for (int o = 16; o > 0; o >>= 1) {
        di += __shfl_xor(di, o, 32);
        dj += __shfl_xor(dj, o, 32);
    }
    if (lane == 0) {
        a.si[g] = di;
        a.sj[g] = dj;
        ((unsigned int*)a.m)[g] = 0xFF800000u;   // -inf bits
        a.s[g] = 0.f;
    }
}

// ---------------------------------------------------------------------------
// Kernel 3: sr[b][t] = rel_emb[b][t] . (Wr[b]^T @ a_r[b])   (10 scalars/block)
// ---------------------------------------------------------------------------
struct SrArgs {
    const float* Wr;     // [NB][256][200]
    const float* A;      // [NB][768]
    const float* rel;    // [NB][10][200]
    float*       sr;     // [NB][16]
};

__global__ __launch_bounds__(256) void sr_kernel(SrArgs a) {
    int b = blockIdx.x;
    int t = threadIdx.x;
    __shared__ float u[RDIM];
    const float* Wr = a.Wr + (size_t)b * DIM * RDIM;
    const float* ar = a.A + (size_t)b * (3 * DIM) + 2 * DIM;
    if (t < RDIM) {
        float acc = 0.f;
        for (int d = 0; d < DIM; ++d) acc += Wr[d * RDIM + t] * ar[d];
        u[t] = acc;
    }
    __syncthreads();
    if (t < NUM_ET) {
        const float* re = a.rel + ((size_t)b * NUM_ET + t) * RDIM;
        float acc = 0.f;
        for (int r = 0; r < RDIM; ++r) acc += re[r] * u[r];
        a.sr[b * 16 + t] = acc;
    }
}

// ---------------------------------------------------------------------------
// Edge kernels: logits + segment max, exp + segment sum, weighted scatter
// ---------------------------------------------------------------------------
struct EdgeArgs {
    const int* ei[3];    // [2][E] per graph
    const int* et[3];    // [E] per graph
    const float* si; const float* sj; const float* sr;
    float* m; float* s; float* ef;
    const float* Wx;
    float* out;
};

__device__ __forceinline__ void atomicMaxF32(float* addr, float val) {
    int bits = __float_as_int(val);
    if (bits >= 0) atomicMax((int*)addr, bits);
    else           atomicMin((unsigned int*)addr, (unsigned int)bits);
}

__global__ __launch_bounds__(256) void edge_max_kernel(EdgeArgs a) {
    int id = blockIdx.x * 256 + threadIdx.x;     // NB*E = 1.92M
    if (id >= NB * N_EDGES) return;
    int b  = id / N_EDGES;
    int e  = id - b * N_EDGES;
    int gr = b >> 1;
    int src = a.ei[gr][e];
    int dst = a.ei[gr][N_EDGES + e];
    int et  = a.et[gr][e];
    float l = a.si[b * N_NODES + dst] + a.sj[b * N_NODES + src] + a.sr[b * 16 + et];
    l = l > 0.f ? l : 0.2f * l;                  // leaky_relu(0.2)
    a.ef[id] = l;
    atomicMaxF32(&a.m[b * N_NODES + dst], l);
}

__global__ __launch_bounds__(256) void edge_exp_kernel(EdgeArgs a) {
    int id = blockIdx.x * 256 + threadIdx.x;
    if (id >= NB * N_EDGES) return;
    int b  = id / N_EDGES;
    int e  = id - b * N_EDGES;
    int gr = b >> 1;
    int dst = a.ei[gr][N_EDGES + e];
    float ev = __expf(a.ef[id] - a.m[b * N_NODES + dst]);
    a.ef[id] = ev;
    atomicAdd(&a.s[b * N_NODES + dst], ev);
}

__global__ __launch_bounds__(256) void edge_scatter_kernel(EdgeArgs a) {
    int g    = blockIdx.x * 8 + (threadIdx.x >> 5);   // one wave per edge
    int lane = threadIdx.x & 31;
    if (g >= NB * N_EDGES) return;
    int b  = g / N_EDGES;
    int e  = g - b * N_EDGES;
    int gr = b >> 1;
    int src = a.ei[gr][e];
    int dst = a.ei[gr][N_EDGES + e];
    float coef = a.ef[g] / a.s[b * N_NODES + dst];
    const float* wrow = a.Wx  + ((size_t)b * N_NODES + src) * DIM;
    float*       orow = a.out + ((size_t)b * N_NODES + dst) * DIM;
    #pragma unroll
    for (int rep = 0; rep < 2; ++rep) {
        int c = rep * 128 + lane * 4;
        float4 v = *(const float4*)(wrow + c);
        atomicAdd(&orow[c + 0], v.x * coef);
        atomicAdd(&orow[c + 1], v.y * coef);
        atomicAdd(&orow[c + 2], v.z * coef);
        atomicAdd(&orow[c + 3], v.w * coef);
    }
}

// ---------------------------------------------------------------------------
// Kernel 7: out = leaky_relu(elu(agg + res), 0.01) elementwise, in place
// ---------------------------------------------------------------------------
__global__ __launch_bounds__(256) void final_act_kernel(float* out) {
    size_t i = ((size_t)blockIdx.x * 256 + threadIdx.x) * 4;
    float4 v = *(float4*)(out + i);
    v.x = v.x > 0.f ? v.x : 0.01f * (__expf(v.x) - 1.f);
    v.y = v.y > 0.f ? v.y : 0.01f * (__expf(v.y) - 1.f);
    v.z = v.z > 0.f ? v.z : 0.01f * (__expf(v.z) - 1.f);
    v.w = v.w > 0.f ? v.w : 0.01f * (__expf(v.w) - 1.f);
    *(float4*)(out + i) = v;
}

// ---------------------------------------------------------------------------
extern "C" void kernel_launch(void* const* d_in, const int* in_sizes, int n_in,
                              void* d_out, int out_size, void* d_ws, size_t ws_size,
                              hipStream_t stream) {
    (void)in_sizes; (void)n_in; (void)out_size; (void)ws_size;

    char* ws = (char*)d_ws;
    const size_t szWx = (size_t)NB * N_NODES * DIM * sizeof(float);  // 122.88 MB
    const size_t szNd = (size_t)NB * N_NODES * sizeof(float);        // 0.48 MB
    float* Wx = (float*)(ws);
    float* si = (float*)(ws + szWx);
    float* sj = (float*)(ws + szWx + szNd);
    float* m  = (float*)(ws + szWx + 2 * szNd);
    float* s  = (float*)(ws + szWx + 3 * szNd);
    float* sr = (float*)(ws + szWx + 4 * szNd);
    float* ef = (float*)(ws + szWx + 4 * szNd + 512);

    GemmArgs ga;
    for (int i = 0; i < NB; ++i) ga.x[i] = (const float*)d_in[i];
    ga.W    = (const float*)d_in[12];
    ga.Wres = (const float*)d_in[15];
    ga.Wx   = Wx;
    ga.out  = (float*)d_out;
    const int ldsBytes = TILEROWS * LDSSTRIDE * (int)sizeof(float);  // 83200
    gemm_wmma_kernel<<<dim3(N_NODES / TILEROWS, NB, 2), 256, ldsBytes, stream>>>(ga);

    NodeArgs na;
    na.Wx = Wx; na.A = (const float*)d_in[14];
    na.si = si; na.sj = sj; na.m = m; na.s = s;
    node_scalar_kernel<<<(NB * N_NODES) / 8, 256, 0, stream>>>(na);

    SrArgs sa;
    sa.Wr = (const float*)d_in[13];
    sa.A  = (const float*)d_in[14];
    sa.rel = (const float*)d_in[16];
    sa.sr = sr;
    sr_kernel<<<NB, 256, 0, stream>>>(sa);

    EdgeArgs ea;
    ea.ei[0] = (const int*)d_in[6];  ea.et[0] = (const int*)d_in[7];
    ea.ei[1] = (const int*)d_in[8];  ea.et[1] = (const int*)d_in[9];
    ea.ei[2] = (const int*)d_in[10]; ea.et[2] = (const int*)d_in[11];
    ea.si = si; ea.sj = sj; ea.sr = sr;
    ea.m = m; ea.s = s; ea.ef = ef;
    ea.Wx = Wx; ea.out = (float*)d_out;

    const int nE = NB * N_EDGES;
    edge_max_kernel<<<(nE + 255) / 256, 256, 0, stream>>>(ea);
    edge_exp_kernel<<<(nE + 255) / 256, 256, 0, stream>>>(ea);
    edge_scatter_kernel<<<nE / 8, 256, 0, stream>>>(ea);

    final_act_kernel<<<(NB * N_NODES * DIM / 4) / 256, 256, 0, stream>>>((float*)d_out);
}